// Xonv2D_61143154426323
// MI455X (gfx1250) — compile-verified
//
#include <hip/hip_runtime.h>

// Xonv2D: per-location conv == 4096 independent GEMMs (M=32, N=32, K=144), fp32.
// Memory-bound (~100 MB @ 23.3 TB/s => ~4.3us floor), so use exact fp32 WMMA
// (V_WMMA_F32_16X16X4_F32) and spend the kernel on clean data movement.

typedef __attribute__((ext_vector_type(2))) float v2f;
typedef __attribute__((ext_vector_type(8))) float v8f;

#define HW_      64
#define NB_      32      // batch
#define CIN_     16
#define COUT_    32
#define KP_      144     // CIN * 3 * 3  (K dim of the per-location GEMM)
#define LDSS_    148     // padded LDS row stride in dwords: 16B-aligned rows,
                         // bank-conflict-free b64 fragment loads (20*i mod 64 distinct)

__global__ __launch_bounds__(128) void xonv2d_wmma_f32(
    const float* __restrict__ x,      // (32,16,64,64)
    const float* __restrict__ wts,    // (64,64,32,16,3,3) -> per-loc [o][p] contiguous
    const float* __restrict__ bias,   // (64,64,32)
    float* __restrict__ out)          // (32,32,64,64)
{
    __shared__ float Wlds[COUT_ * LDSS_];  // weights, [o][p] rows
    __shared__ float Alds[NB_   * LDSS_];  // im2col patches, [b][p] rows

    const int loc = blockIdx.x;            // h*64 + w
    const int h   = loc >> 6;
    const int w   = loc & 63;
    const int tid = threadIdx.x;           // 0..127

    // ---- Stage per-location weights: 4608 contiguous floats, verbatim [o][p].
    // 1152 float4 loads, 9 per thread; LDS rows are 16B aligned (148*4 = 592 = 37*16).
    {
        const float4* __restrict__ src =
            (const float4*)(wts + (size_t)loc * (COUT_ * KP_));
        #pragma unroll
        for (int j = 0; j < 9; ++j) {
            const int g4 = tid + j * 128;       // 0..1151
            const int o  = g4 / 36;             // 36 float4 per 144-float row
            const int p4 = g4 - o * 36;
            float4 v = src[g4];
            *(float4*)(&Wlds[o * LDSS_ + p4 * 4]) = v;
        }
    }

    // ---- Build im2col patch tile A[b][p], zero-padded at image borders.
    // p = c*9 + r*3 + s (channel-slowest, matches reference patch ordering).
    {
        #pragma unroll
        for (int j = 0; j < 36; ++j) {
            const int i  = tid + j * 128;       // 0..4607
            const int b  = i / KP_;
            const int p  = i - b * KP_;
            const int c  = p / 9;
            const int rs = p - c * 9;
            const int r  = rs / 3;
            const int s  = rs - r * 3;
            const int yy = h + r - 1;
            const int xx = w + s - 1;
            float v = 0.0f;
            if ((unsigned)yy < 64u && (unsigned)xx < 64u)
                v = x[(((b * CIN_ + c) * HW_) + yy) * HW_ + xx];
            Alds[b * LDSS_ + p] = v;
        }
    }

    __syncthreads();

    // ---- 4 waves, one 16x16 tile each of the 32x32 output.
    const int wave   = tid >> 5;
    const int lane   = tid & 31;
    const int m_tile = wave & 1;       // batch-tile
    const int n_tile = wave >> 1;      // cout-tile
    const int rm     = lane & 15;
    const int half   = lane >> 4;

    // fp32 WMMA fragment layout (ISA 7.12.2):
    //   A 16x4: lane(half) holds (M = rm, K = k0 + 2*half + {0,1})  -> float2
    //   B 4x16: lane(half) holds (K = k0 + 2*half + {0,1}, N = rm)  -> float2
    const float* __restrict__ arow = &Alds[(m_tile * 16 + rm) * LDSS_ + 2 * half];
    const float* __restrict__ brow = &Wlds[(n_tile * 16 + rm) * LDSS_ + 2 * half];

    v8f acc = {};
    #pragma unroll
    for (int k0 = 0; k0 < KP_; k0 += 4) {
        v2f afrag = *(const v2f*)(arow + k0);
        v2f bfrag = *(const v2f*)(brow + k0);
        acc = __builtin_amdgcn_wmma_f32_16x16x4_f32(
            /*neg_a=*/false, afrag, /*neg_b=*/false, bfrag,
            /*c_mod=*/(short)0, acc, /*reuse_a=*/false, /*reuse_b=*/false);
    }

    // ---- Epilogue: C/D layout -> VGPR r: M = r + 8*half, N = rm (fixed o per lane).
    const int   o  = n_tile * 16 + rm;
    const float bv = bias[loc * COUT_ + o];
    #pragma unroll
    for (int r = 0; r < 8; ++r) {
        const int b = m_tile * 16 + half * 8 + r;
        out[(size_t)(b * COUT_ + o) * (HW_ * HW_) + loc] = acc[r] + bv;
    }
}

extern "C" void kernel_launch(void* const* d_in, const int* in_sizes, int n_in,
                              void* d_out, int out_size, void* d_ws, size_t ws_size,
                              hipStream_t stream) {
    (void)in_sizes; (void)n_in; (void)d_ws; (void)ws_size; (void)out_size;
    const float* x    = (const float*)d_in[0];
    const float* wts  = (const float*)d_in[1];
    const float* bias = (const float*)d_in[2];
    float* out        = (float*)d_out;

    dim3 grid(HW_ * HW_);   // 4096 locations
    dim3 block(128);        // 4 waves
    xonv2d_wmma_f32<<<grid, block, 0, stream>>>(x, wts, bias, out);
}